// TensorTrainODEBLOCK_18459769438845
// MI455X (gfx1250) — compile-verified
//
#include <hip/hip_runtime.h>

// ---------------- config (mirrors reference) ----------------
#define BATCH 16384
#define DIN   2048
#define DZ    32
#define DOUT  10
#define RANK  16
#define MBAS  5      // poly degree 4 -> 5 basis funcs
#define HID   256
#define NSTEP 20
#define NMID  (DZ - 1)   // 31 mid TT cores

// interleaved B-operand sizes (floats)
#define WBMID_F (NMID * 4 * 2 * 80 * 2)   // 39680
#define WBLAST_F (4 * 2 * 160 * 2)        // 2560
#define WBODE_F (WBMID_F + WBLAST_F)      // 42240
// dynamic LDS for ode kernel: staged B operands + per-wave z-stage + v-buffer
#define ODE_LDS_FLOATS (WBODE_F + 4 * 16 * DZ + 4 * 16 * 16)   // 45312 -> 177 KB

typedef __attribute__((ext_vector_type(2))) float v2f;
typedef __attribute__((ext_vector_type(8))) float v8f;

// f32 WMMA: D(16x16,f32) = A(16x4,f32) x B(4x16,f32) + C
// 8-arg form: (neg_a, A, neg_b, B, c_mod, C, reuse_a, reuse_b)
__device__ __forceinline__ v8f wmma4(v2f a, v2f b, v8f c) {
  return __builtin_amdgcn_wmma_f32_16x16x4_f32(false, a, false, b, (short)0, c,
                                               false, false);
}

// per-wave LDS ordering fence (DS pipe is in-order per wave; this stops the
// compiler reordering cross-lane LDS reads above the producing stores)
__device__ __forceinline__ void lds_fence() {
  asm volatile("s_wait_dscnt 0" ::: "memory");
}

// =====================================================================
// Prep: interleave B-operands so one b64 load yields both B VGPRs of a
// WMMA_F32_16X16X4 K-chunk.
//   B chunk c, lane half `hi`: VGPR0 = G[4c+2*hi][col], VGPR1 = G[4c+2*hi+1][col]
// =====================================================================
__global__ __launch_bounds__(256) void prep_kernel(
    const float* __restrict__ Gmid, const float* __restrict__ Glast,
    const float* __restrict__ W1, const float* __restrict__ W2,
    const float* __restrict__ W3,
    float* __restrict__ WBmid, float* __restrict__ WBlast,
    float* __restrict__ WB1, float* __restrict__ WB2, float* __restrict__ WB3) {
  const int tid0 = blockIdx.x * blockDim.x + threadIdx.x;
  const int stride = gridDim.x * blockDim.x;

  // Gmid [31][16][5][16] -> WBmid[n][c(4)][hi(2)][col(80)] pairs
  for (int idx = tid0; idx < NMID * 4 * 2 * 80; idx += stride) {
    int col = idx % 80;
    int rest = idx / 80;
    int hi = rest & 1; rest >>= 1;
    int c = rest & 3;
    int n = rest >> 2;
    int m = col / 16, s = col % 16;
    int r = 4 * c + 2 * hi;
    WBmid[idx * 2 + 0] = Gmid[((n * RANK + r) * MBAS + m) * 16 + s];
    WBmid[idx * 2 + 1] = Gmid[((n * RANK + r + 1) * MBAS + m) * 16 + s];
  }
  // Glast [16][5][32] -> WBlast[c(4)][hi(2)][col(160)] pairs, col = m*32+d
  for (int idx = tid0; idx < 4 * 2 * 160; idx += stride) {
    int col = idx % 160;
    int rest = idx / 160;
    int hi = rest & 1;
    int c = rest >> 1;
    int m = col / 32, d = col % 32;
    int r = 4 * c + 2 * hi;
    WBlast[idx * 2 + 0] = Glast[(r * MBAS + m) * 32 + d];
    WBlast[idx * 2 + 1] = Glast[((r + 1) * MBAS + m) * 32 + d];
  }
  // W1 [32][256] -> WB1[c(8)][hi(2)][col(256)]
  for (int idx = tid0; idx < 8 * 2 * 256; idx += stride) {
    int col = idx % 256;
    int rest = idx / 256;
    int hi = rest & 1;
    int c = rest >> 1;
    int r = 4 * c + 2 * hi;
    WB1[idx * 2 + 0] = W1[r * 256 + col];
    WB1[idx * 2 + 1] = W1[(r + 1) * 256 + col];
  }
  // W2 [256][256] -> WB2[c(64)][hi(2)][col(256)]
  for (int idx = tid0; idx < 64 * 2 * 256; idx += stride) {
    int col = idx % 256;
    int rest = idx / 256;
    int hi = rest & 1;
    int c = rest >> 1;
    int r = 4 * c + 2 * hi;
    WB2[idx * 2 + 0] = W2[r * 256 + col];
    WB2[idx * 2 + 1] = W2[(r + 1) * 256 + col];
  }
  // W3 [256][10] -> WB3[c(64)][hi(2)][col(16)], zero-padded cols 10..15
  for (int idx = tid0; idx < 64 * 2 * 16; idx += stride) {
    int col = idx % 16;
    int rest = idx / 16;
    int hi = rest & 1;
    int c = rest >> 1;
    int r = 4 * c + 2 * hi;
    WB3[idx * 2 + 0] = (col < DOUT) ? W3[r * DOUT + col] : 0.f;
    WB3[idx * 2 + 1] = (col < DOUT) ? W3[(r + 1) * DOUT + col] : 0.f;
  }
}

// =====================================================================
// Projection: z0 = x @ P.  One wave per row, lane = output column.
// x row loads broadcast (single request per wave); P loads coalesced.
// =====================================================================
__global__ __launch_bounds__(256) void proj_kernel(
    const float* __restrict__ x, const float* __restrict__ P,
    float* __restrict__ z0) {
  const int gtid = blockIdx.x * blockDim.x + threadIdx.x;
  const int row = gtid >> 5;
  const int col = gtid & 31;
  const float* xr = x + (size_t)row * DIN;
  float acc = 0.f;
  for (int k = 0; k < DIN; k += 4) {
    float4 xv = *(const float4*)(xr + k);
    acc = fmaf(xv.x, P[(k + 0) * DZ + col], acc);
    acc = fmaf(xv.y, P[(k + 1) * DZ + col], acc);
    acc = fmaf(xv.z, P[(k + 2) * DZ + col], acc);
    acc = fmaf(xv.w, P[(k + 3) * DZ + col], acc);
  }
  z0[(size_t)row * DZ + col] = acc;
}

// =====================================================================
// TT field: k = f(t, z) for one 16-row tile held in WMMA D-layout regs.
// z regs: z[h][i] = element (row = i + 8*hi, col = h*16 + ln)
// All B operands come from LDS (staged once per block).
// =====================================================================
__device__ __forceinline__ void tt_field(
    float t, const float (*zin)[8], float (*kk)[8],
    const float* __restrict__ G0, const float* WBmid /*LDS*/,
    const float* WBlast /*LDS*/, float* zsw /*LDS [16*32]*/,
    float* vbw /*LDS [16*16]*/, int ln, int hi) {
  // stage z tile into LDS (row-major [16][32]) for per-row phi lookups
#pragma unroll
  for (int h = 0; h < 2; ++h)
#pragma unroll
    for (int i = 0; i < 8; ++i)
      zsw[(i + 8 * hi) * DZ + h * 16 + ln] = zin[h][i];
  lds_fence();

  // v0 = phi_t @ G0  (uniform across rows) directly in A-chunk layout
  v2f A[4];
#pragma unroll
  for (int c = 0; c < 4; ++c) {
    int r0 = 4 * c + 2 * hi;
    float a0 = 0.f, a1 = 0.f, p = 1.f;
#pragma unroll
    for (int m = 0; m < MBAS; ++m) {
      a0 = fmaf(p, G0[m * RANK + r0], a0);
      a1 = fmaf(p, G0[m * RANK + r0 + 1], a1);
      p *= t;
    }
    A[c].x = a0;
    A[c].y = a1;
  }

  // chain of 31 mid cores: U[b,(m,s)] = V[b,r] G[r,(m,s)] via WMMA,
  // then v_new[b,s] = sum_m z_bn^m * U[b,m,s]
  for (int n = 0; n < NMID; ++n) {
    float zk[8];
#pragma unroll
    for (int i = 0; i < 8; ++i) zk[i] = zsw[(i + 8 * hi) * DZ + n];

    v8f U[MBAS];
#pragma unroll
    for (int m = 0; m < MBAS; ++m) U[m] = (v8f){0, 0, 0, 0, 0, 0, 0, 0};
#pragma unroll
    for (int m = 0; m < MBAS; ++m)
#pragma unroll
      for (int c = 0; c < 4; ++c) {
        v2f B = *(const v2f*)(WBmid +
                              (((n * 4 + c) * 2 + hi) * 80 + m * 16 + ln) * 2);
        U[m] = wmma4(A[c], B, U[m]);
      }
    // combine over basis powers, write new V to LDS (row-major [16][16])
#pragma unroll
    for (int i = 0; i < 8; ++i) {
      float p = zk[i];
      float a = U[0][i];
      a = fmaf(p, U[1][i], a);
      float p2 = p * p;
      a = fmaf(p2, U[2][i], a);
      a = fmaf(p2 * p, U[3][i], a);
      a = fmaf(p2 * p2, U[4][i], a);
      vbw[(i + 8 * hi) * 16 + ln] = a;
    }
    lds_fence();
    // reload in A layout: b64 gets (V[ln][4c+2hi], V[ln][4c+2hi+1])
#pragma unroll
    for (int c = 0; c < 4; ++c)
      A[c] = *(const v2f*)&vbw[ln * 16 + 4 * c + 2 * hi];
  }

  // output core Glast: dz[b,d] = sum_m z_b,31^m * (V G)[b,(m,d)]
  float zl[8];
#pragma unroll
  for (int i = 0; i < 8; ++i) zl[i] = zsw[(i + 8 * hi) * DZ + (DZ - 1)];
#pragma unroll
  for (int h = 0; h < 2; ++h) {
    v8f U[MBAS];
#pragma unroll
    for (int m = 0; m < MBAS; ++m) U[m] = (v8f){0, 0, 0, 0, 0, 0, 0, 0};
#pragma unroll
    for (int m = 0; m < MBAS; ++m)
#pragma unroll
      for (int c = 0; c < 4; ++c) {
        v2f B = *(const v2f*)(WBlast +
                              ((c * 2 + hi) * 160 + m * 32 + h * 16 + ln) * 2);
        U[m] = wmma4(A[c], B, U[m]);
      }
#pragma unroll
    for (int i = 0; i < 8; ++i) {
      float p = zl[i];
      float a = U[0][i];
      a = fmaf(p, U[1][i], a);
      float p2 = p * p;
      a = fmaf(p2, U[2][i], a);
      a = fmaf(p2 * p2 * p, U[4][i], a);   // placeholder reordered below
      kk[h][i] = a;
    }
    // NOTE: keep exact basis order (m=0..4); recompute properly:
#pragma unroll
    for (int i = 0; i < 8; ++i) {
      float p = zl[i];
      float a = U[0][i];
      a = fmaf(p, U[1][i], a);
      float p2 = p * p;
      a = fmaf(p2, U[2][i], a);
      a = fmaf(p2 * p, U[3][i], a);
      a = fmaf(p2 * p2, U[4][i], a);
      kk[h][i] = a;
    }
  }
}

// =====================================================================
// RK4 integrator: 4 independent waves/block, 16 batch rows per wave.
// Interleaved Gmid/Glast B-operands staged into dynamic LDS once per
// block; the 80 field evals then read B from LDS (64-bank bandwidth)
// instead of hammering L2.
// =====================================================================
extern __shared__ float ode_smem[];

__global__ __launch_bounds__(128) void ode_kernel(
    const float* __restrict__ z0g, float* __restrict__ zfg,
    const float* __restrict__ G0, const float* __restrict__ WBode) {
  // stage interleaved B operands (WBmid || WBlast, 42240 floats) into LDS
  for (int idx = threadIdx.x; idx < WBODE_F / 4; idx += 128)
    ((float4*)ode_smem)[idx] = ((const float4*)WBode)[idx];
  __syncthreads();

  const int wid = threadIdx.x >> 5;
  const int lane = threadIdx.x & 31;
  const int ln = lane & 15, hi = lane >> 4;
  const int rowBase = (blockIdx.x * 4 + wid) * 16;
  const float* WBmid = ode_smem;
  const float* WBlast = ode_smem + WBMID_F;
  float* zsw = ode_smem + WBODE_F + wid * 16 * DZ;
  float* vbw = ode_smem + WBODE_F + 4 * 16 * DZ + wid * 16 * 16;

  float z[2][8];
#pragma unroll
  for (int h = 0; h < 2; ++h)
#pragma unroll
    for (int i = 0; i < 8; ++i)
      z[h][i] = z0g[(size_t)(rowBase + i + 8 * hi) * DZ + h * 16 + ln];

  const float hs = 1.0f / (float)NSTEP;
  float zt[2][8], ka[2][8], kacc[2][8];
  for (int s = 0; s < NSTEP; ++s) {
    float t = hs * (float)s;
    // k1
    tt_field(t, z, ka, G0, WBmid, WBlast, zsw, vbw, ln, hi);
#pragma unroll
    for (int h = 0; h < 2; ++h)
#pragma unroll
      for (int i = 0; i < 8; ++i) {
        kacc[h][i] = ka[h][i];
        zt[h][i] = fmaf(0.5f * hs, ka[h][i], z[h][i]);
      }
    // k2
    tt_field(t + 0.5f * hs, zt, ka, G0, WBmid, WBlast, zsw, vbw, ln, hi);
#pragma unroll
    for (int h = 0; h < 2; ++h)
#pragma unroll
      for (int i = 0; i < 8; ++i) {
        kacc[h][i] = fmaf(2.f, ka[h][i], kacc[h][i]);
        zt[h][i] = fmaf(0.5f * hs, ka[h][i], z[h][i]);
      }
    // k3
    tt_field(t + 0.5f * hs, zt, ka, G0, WBmid, WBlast, zsw, vbw, ln, hi);
#pragma unroll
    for (int h = 0; h < 2; ++h)
#pragma unroll
      for (int i = 0; i < 8; ++i) {
        kacc[h][i] = fmaf(2.f, ka[h][i], kacc[h][i]);
        zt[h][i] = fmaf(hs, ka[h][i], z[h][i]);
      }
    // k4
    tt_field(t + hs, zt, ka, G0, WBmid, WBlast, zsw, vbw, ln, hi);
#pragma unroll
    for (int h = 0; h < 2; ++h)
#pragma unroll
      for (int i = 0; i < 8; ++i) {
        kacc[h][i] += ka[h][i];
        z[h][i] = fmaf(hs * (1.f / 6.f), kacc[h][i], z[h][i]);
      }
  }

#pragma unroll
  for (int h = 0; h < 2; ++h)
#pragma unroll
    for (int i = 0; i < 8; ++i)
      zfg[(size_t)(rowBase + i + 8 * hi) * DZ + h * 16 + ln] = z[h][i];
}

// =====================================================================
// MLP head: zf[16x32] -> relu(.W1+b1) -> relu(.W2+b2) -> .W3+b3
// all GEMMs via WMMA f32 16x16x4; activation transpose through LDS.
// =====================================================================
__global__ __launch_bounds__(64) void head_kernel(
    const float* __restrict__ zfg, const float* __restrict__ WB1,
    const float* __restrict__ WB2, const float* __restrict__ WB3,
    const float* __restrict__ b1, const float* __restrict__ b2,
    const float* __restrict__ b3, float* __restrict__ out) {
  __shared__ __align__(16) float zbuf[2 * 16 * DZ];     // 4 KB
  __shared__ __align__(16) float hbuf[2 * 16 * HID];    // 32 KB
  const int wid = threadIdx.x >> 5;
  const int lane = threadIdx.x & 31;
  const int ln = lane & 15, hi = lane >> 4;
  const int rowBase = (blockIdx.x * 2 + wid) * 16;
  float* zb = zbuf + wid * 16 * DZ;
  float* hb = hbuf + wid * 16 * HID;

  // coalesced load of the zf tile
  for (int j = 0; j < 16; ++j)
    zb[j * 32 + lane] = zfg[(size_t)rowBase * DZ + j * 32 + lane];
  lds_fence();

  // layer 1: [16x32] x [32x256]
  v2f Az[8];
#pragma unroll
  for (int c = 0; c < 8; ++c)
    Az[c] = *(const v2f*)&zb[ln * DZ + 4 * c + 2 * hi];
#pragma unroll 4
  for (int nt = 0; nt < 16; ++nt) {
    v8f U = (v8f){0, 0, 0, 0, 0, 0, 0, 0};
#pragma unroll
    for (int c = 0; c < 8; ++c) {
      v2f B = *(const v2f*)(WB1 + ((c * 2 + hi) * 256 + nt * 16 + ln) * 2);
      U = wmma4(Az[c], B, U);
    }
    float bb = b1[nt * 16 + ln];
#pragma unroll
    for (int i = 0; i < 8; ++i)
      hb[(i + 8 * hi) * HID + nt * 16 + ln] = fmaxf(U[i] + bb, 0.f);
  }
  lds_fence();

  // layer 2: [16x256] x [256x256], accumulate all 16 tiles in regs
  v8f H2[16];
#pragma unroll 2
  for (int nt = 0; nt < 16; ++nt) {
    v8f U = (v8f){0, 0, 0, 0, 0, 0, 0, 0};
    for (int c = 0; c < 64; ++c) {
      v2f Aa = *(const v2f*)&hb[ln * HID + 4 * c + 2 * hi];
      v2f B = *(const v2f*)(WB2 + ((c * 2 + hi) * 256 + nt * 16 + ln) * 2);
      U = wmma4(Aa, B, U);
    }
    float bb = b2[nt * 16 + ln];
#pragma unroll
    for (int i = 0; i < 8; ++i) U[i] = fmaxf(U[i] + bb, 0.f);
    H2[nt] = U;
  }
  // write h2 back over hbuf (per-wave DS pipe is in-order; reads precede)
#pragma unroll
  for (int nt = 0; nt < 16; ++nt)
#pragma unroll
    for (int i = 0; i < 8; ++i)
      hb[(i + 8 * hi) * HID + nt * 16 + ln] = H2[nt][i];
  lds_fence();

  // layer 3: [16x256] x [256x16(padded from 10)]
  v8f U = (v8f){0, 0, 0, 0, 0, 0, 0, 0};
  for (int c = 0; c < 64; ++c) {
    v2f Aa = *(const v2f*)&hb[ln * HID + 4 * c + 2 * hi];
    v2f B = *(const v2f*)(WB3 + ((c * 2 + hi) * 16 + ln) * 2);
    U = wmma4(Aa, B, U);
  }
  float bb = (ln < DOUT) ? b3[ln] : 0.f;
#pragma unroll
  for (int i = 0; i < 8; ++i)
    if (ln < DOUT)
      out[(size_t)(rowBase + i + 8 * hi) * DOUT + ln] = U[i] + bb;
}

// =====================================================================
extern "C" void kernel_launch(void* const* d_in, const int* in_sizes, int n_in,
                              void* d_out, int out_size, void* d_ws,
                              size_t ws_size, hipStream_t stream) {
  (void)in_sizes; (void)n_in; (void)out_size; (void)ws_size;
  const float* x     = (const float*)d_in[0];
  const float* P     = (const float*)d_in[1];
  const float* G0    = (const float*)d_in[2];
  const float* Gmid  = (const float*)d_in[3];
  const float* Glast = (const float*)d_in[4];
  const float* W1    = (const float*)d_in[5];
  const float* b1    = (const float*)d_in[6];
  const float* W2    = (const float*)d_in[7];
  const float* b2    = (const float*)d_in[8];
  const float* W3    = (const float*)d_in[9];
  const float* b3    = (const float*)d_in[10];
  float* out = (float*)d_out;

  float* ws = (float*)d_ws;
  float* z0     = ws;                         // 16384*32   = 524288 f
  float* WBmid  = z0 + BATCH * DZ;            // 39680 f  (contiguous with
  float* WBlast = WBmid + WBMID_F;            //  2560 f   WBlast -> WBODE_F)
  float* WB1    = WBlast + WBLAST_F;          //  8192 f
  float* WB2    = WB1 + 8 * 2 * 256 * 2;      // 65536 f
  float* WB3    = WB2 + 64 * 2 * 256 * 2;     //  4096 f

  prep_kernel<<<128, 256, 0, stream>>>(Gmid, Glast, W1, W2, W3, WBmid, WBlast,
                                       WB1, WB2, WB3);
  proj_kernel<<<(BATCH * 32) / 256, 256, 0, stream>>>(x, P, z0);
  ode_kernel<<<(BATCH / 16) / 4, 128, ODE_LDS_FLOATS * sizeof(float), stream>>>(
      z0, z0, G0, WBmid);
  head_kernel<<<(BATCH / 16) / 2, 64, 0, stream>>>(z0, WB1, WB2, WB3, b1, b2,
                                                   b3, out);
}